// GCNEncoder_26276609917208
// MI455X (gfx1250) — compile-verified
//
#include <hip/hip_runtime.h>

typedef float v2f __attribute__((ext_vector_type(2)));
typedef float v8f __attribute__((ext_vector_type(8)));

// ---------------------------------------------------------------------------
// Degree / normalization
// ---------------------------------------------------------------------------
__global__ void k_init_deg(float* __restrict__ deg, int n) {
  int i = blockIdx.x * blockDim.x + threadIdx.x;
  if (i < n) deg[i] = 1.0f;  // self-loop contributes 1
}

__global__ void k_deg_accum(const long long* __restrict__ ei, float* __restrict__ deg, int E) {
  int e = blockIdx.x * blockDim.x + threadIdx.x;
  if (e < E) atomicAdd(&deg[(int)ei[E + e]], 1.0f);  // dst row of edge_index
}

__global__ void k_dinv(float* __restrict__ deg, int n) {
  int i = blockIdx.x * blockDim.x + threadIdx.x;
  if (i < n) {
    float d = deg[i];
    deg[i] = (d > 0.0f) ? rsqrtf(d) : 0.0f;  // in-place: deg -> dinv
  }
}

// ---------------------------------------------------------------------------
// GEMM1: h[N,32] = x[N,4] @ W1[4,32]  via V_WMMA_F32_16X16X4_F32.
// Fused epilogue: a[m,c] = b1[c] + h[m,c]*dinv[m]^2  (self-loop msg + bias).
// One wave = one 16-row tile, two 16x16 output tiles (cols 0-15, 16-31).
// ---------------------------------------------------------------------------
__global__ void k_gemm1(const float* __restrict__ x, const float* __restrict__ W1,
                        const float* __restrict__ dinv, const float* __restrict__ b1,
                        float* __restrict__ h, float* __restrict__ a, int n) {
  int wave = (int)((blockIdx.x * blockDim.x + threadIdx.x) >> 5);
  int lane = threadIdx.x & 31;
  int tiles = (n + 15) >> 4;
  if (wave >= tiles) return;          // wave-uniform: EXEC all-1s at WMMA

  int row0 = wave << 4;
  int r    = lane & 15;               // row (A) / col (B,D) within tile
  int kb   = (lane >> 4) << 1;        // lanes 0-15: K=0,1 ; lanes 16-31: K=2,3

  int rr = row0 + r; if (rr >= n) rr = n - 1;   // clamped load (stores guarded)

  v2f av = *(const v2f*)(x + rr * 4 + kb);               // 8B-aligned pair
  v2f b0; b0.x = W1[kb * 32 + r];       b0.y = W1[(kb + 1) * 32 + r];
  v2f b1v; b1v.x = W1[kb * 32 + 16 + r]; b1v.y = W1[(kb + 1) * 32 + 16 + r];

  v8f c0 = {}; v8f c1 = {};
  c0 = __builtin_amdgcn_wmma_f32_16x16x4_f32(false, av, false, b0, (short)0, c0, false, false);
  c1 = __builtin_amdgcn_wmma_f32_16x16x4_f32(false, av, false, b1v, (short)0, c1, false, false);

  float bias0 = b1[r];
  float bias1 = b1[16 + r];
  int mbase = row0 + ((lane >> 4) << 3);  // lanes>=16 hold M=v+8

  if (row0 + 16 <= n) {                   // wave-uniform fast path (all tiles but last)
#pragma unroll
    for (int v = 0; v < 8; ++v) {
      int m = mbase + v;
      float di = dinv[m];                 // half-wave broadcast, L2-hot
      float d2 = di * di;
      h[m * 32 + r]      = c0[v];
      h[m * 32 + 16 + r] = c1[v];
      a[m * 32 + r]      = bias0 + c0[v] * d2;
      a[m * 32 + 16 + r] = bias1 + c1[v] * d2;
    }
  } else {
#pragma unroll
    for (int v = 0; v < 8; ++v) {
      int m = mbase + v;
      if (m < n) {
        float di = dinv[m];
        float d2 = di * di;
        h[m * 32 + r]      = c0[v];
        h[m * 32 + 16 + r] = c1[v];
        a[m * 32 + r]      = bias0 + c0[v] * d2;
        a[m * 32 + 16 + r] = bias1 + c1[v] * d2;
      }
    }
  }
}

// edge scatter, layer 1: 8 threads per edge, 4 channels each (float4 gather)
__global__ void k_scatter1(const long long* __restrict__ ei, const float* __restrict__ dinv,
                           const float* __restrict__ h, float* __restrict__ a, int E) {
  int t = blockIdx.x * blockDim.x + threadIdx.x;
  if (t >= E * 8) return;
  int e = t >> 3, g = (t & 7) << 2;
  int s = (int)ei[e], d = (int)ei[E + e];
  float nrm = dinv[s] * dinv[d];
  float4 hv = *(const float4*)(h + s * 32 + g);
  float* ap = a + d * 32 + g;
  atomicAdd(ap + 0, hv.x * nrm);
  atomicAdd(ap + 1, hv.y * nrm);
  atomicAdd(ap + 2, hv.z * nrm);
  atomicAdd(ap + 3, hv.w * nrm);
}

// ---------------------------------------------------------------------------
// GEMM2: h2[N,16] = relu(a[N,32]) @ W2[32,16], ReLU fused into A load.
// K=32 -> chain of 8 WMMA f32 16x16x4 accumulating through C.
// Fused epilogue: out[m,c] = b2[c] + h2[m,c]*dinv[m]^2.
// ---------------------------------------------------------------------------
__global__ void k_gemm2(const float* __restrict__ a, const float* __restrict__ W2,
                        const float* __restrict__ dinv, const float* __restrict__ b2,
                        float* __restrict__ h2, float* __restrict__ out, int n) {
  int wave = (int)((blockIdx.x * blockDim.x + threadIdx.x) >> 5);
  int lane = threadIdx.x & 31;
  int tiles = (n + 15) >> 4;
  if (wave >= tiles) return;

  int row0 = wave << 4;
  int r    = lane & 15;
  int kh   = (lane >> 4) << 1;

  int rr = row0 + r; if (rr >= n) rr = n - 1;

  v8f acc = {};
#pragma unroll
  for (int kk = 0; kk < 8; ++kk) {
    int k = kk * 4 + kh;
    v2f av = *(const v2f*)(a + rr * 32 + k);   // 8B-aligned pair
    av.x = fmaxf(av.x, 0.0f);                  // fused ReLU
    av.y = fmaxf(av.y, 0.0f);
    v2f bv;
    bv.x = W2[k * 16 + r];
    bv.y = W2[(k + 1) * 16 + r];
    acc = __builtin_amdgcn_wmma_f32_16x16x4_f32(false, av, false, bv, (short)0, acc, false, false);
  }

  float bias = b2[r];
  int mbase = row0 + ((lane >> 4) << 3);

  if (row0 + 16 <= n) {                        // wave-uniform fast path
#pragma unroll
    for (int v = 0; v < 8; ++v) {
      int m = mbase + v;
      float di = dinv[m];
      h2[m * 16 + r]  = acc[v];
      out[m * 16 + r] = bias + acc[v] * di * di;
    }
  } else {
#pragma unroll
    for (int v = 0; v < 8; ++v) {
      int m = mbase + v;
      if (m < n) {
        float di = dinv[m];
        h2[m * 16 + r]  = acc[v];
        out[m * 16 + r] = bias + acc[v] * di * di;
      }
    }
  }
}

// edge scatter, layer 2: 4 threads per edge, 4 channels each
__global__ void k_scatter2(const long long* __restrict__ ei, const float* __restrict__ dinv,
                           const float* __restrict__ h2, float* __restrict__ out, int E) {
  int t = blockIdx.x * blockDim.x + threadIdx.x;
  if (t >= E * 4) return;
  int e = t >> 2, g = (t & 3) << 2;
  int s = (int)ei[e], d = (int)ei[E + e];
  float nrm = dinv[s] * dinv[d];
  float4 hv = *(const float4*)(h2 + s * 16 + g);
  float* op = out + d * 16 + g;
  atomicAdd(op + 0, hv.x * nrm);
  atomicAdd(op + 1, hv.y * nrm);
  atomicAdd(op + 2, hv.z * nrm);
  atomicAdd(op + 3, hv.w * nrm);
}

// ---------------------------------------------------------------------------
extern "C" void kernel_launch(void* const* d_in, const int* in_sizes, int n_in,
                              void* d_out, int out_size, void* d_ws, size_t ws_size,
                              hipStream_t stream) {
  const float*     x  = (const float*)d_in[0];
  const long long* ei = (const long long*)d_in[1];   // int64 edge_index (2,E)
  const float*     W1 = (const float*)d_in[2];
  const float*     b1 = (const float*)d_in[3];
  const float*     W2 = (const float*)d_in[4];
  const float*     b2 = (const float*)d_in[5];
  float*           out = (float*)d_out;

  int n = in_sizes[0] / 4;   // nodes (in_c = 4)
  int E = in_sizes[1] / 2;   // edges

  // workspace layout (floats): dinv[n] | h[n*32] | a[n*32] | h2[n*16]
  float* ws   = (float*)d_ws;
  size_t o    = ((size_t)n + 255) & ~(size_t)255;
  float* dinv = ws;
  float* h    = ws + o;
  float* a    = h + (size_t)n * 32;
  float* h2   = a + (size_t)n * 32;

  const int B = 256;               // 8 waves per block (wave32)
  const int wpb = B / 32;
  int tiles = (n + 15) / 16;

  k_init_deg <<<(n + B - 1) / B, B, 0, stream>>>(dinv, n);
  k_deg_accum<<<(E + B - 1) / B, B, 0, stream>>>(ei, dinv, E);
  k_dinv     <<<(n + B - 1) / B, B, 0, stream>>>(dinv, n);

  k_gemm1    <<<(tiles + wpb - 1) / wpb, B, 0, stream>>>(x, W1, dinv, b1, h, a, n);
  k_scatter1 <<<(E * 8 + B - 1) / B, B, 0, stream>>>(ei, dinv, h, a, E);

  k_gemm2    <<<(tiles + wpb - 1) / wpb, B, 0, stream>>>(a, W2, dinv, b2, h2, out, n);
  k_scatter2 <<<(E * 4 + B - 1) / B, B, 0, stream>>>(ei, dinv, h2, out, E);
}